// ConvBlock_7919919693901
// MI455X (gfx1250) — compile-verified
//
#include <hip/hip_runtime.h>
#include <math.h>

typedef __attribute__((ext_vector_type(2))) float v2f;
typedef __attribute__((ext_vector_type(8))) float v8f;

static constexpr int C     = 128;     // C_IN == C_OUT
static constexpr int T_CH  = 512;     // temb channels
static constexpr int E_SEG = 131072;  // number of hyperedges (fixed in reference)

// ---------------------------------------------------------------- utilities

__global__ __launch_bounds__(256) void zero_f32(float* __restrict__ p, size_t n) {
  size_t i = (size_t)blockIdx.x * blockDim.x + threadIdx.x;
  size_t stride = (size_t)gridDim.x * blockDim.x;
  for (; i < n; i += stride) p[i] = 0.0f;
}

// node/hyperedge degree counting (float atomics; arrays pre-zeroed)
__global__ __launch_bounds__(256) void degree_kernel(const int* __restrict__ node_idx,
                                                     const int* __restrict__ edge_idx,
                                                     float* __restrict__ D,
                                                     float* __restrict__ B, int nnz) {
  int i = blockIdx.x * 256 + threadIdx.x;
  if (i < nnz) {
    atomicAdd(&D[node_idx[i]], 1.0f);
    atomicAdd(&B[edge_idx[i]], 1.0f);
  }
}

__global__ __launch_bounds__(256) void recip_kernel(float* __restrict__ d, int n) {
  int i = blockIdx.x * 256 + threadIdx.x;
  if (i < n) {
    float v = d[i];
    d[i] = (v > 0.0f) ? (1.0f / v) : 0.0f;
  }
}

// ------------------------------------------------ dense GEMM: xt = x @ W^T
// fp32 WMMA 16x16x4. One wave computes a 16-row x 128-col strip of xt.
// A fragment (16x4 f32, 2 VGPRs/lane): row M = lane&15; lanes 0-15 hold
//   K+0/K+1, lanes 16-31 hold K+2/K+3 (ISA 7.12.2 "32-bit A-Matrix 16x4").
// B fragment (4x16, B = W^T): col N = lane&15, mirrored K split.
// C/D (8 VGPRs): VGPR v -> row M = v + (lane>=16 ? 8 : 0), col N = lane&15.
__global__ __launch_bounds__(256) void gemm_xt_wmma(const float* __restrict__ x,
                                                    const float* __restrict__ W,
                                                    float* __restrict__ xt, int N) {
  const int wave = threadIdx.x >> 5;        // 0..7  (wave32)
  const int lane = threadIdx.x & 31;
  const int m    = lane & 15;
  const int kh   = (lane >> 4) ? 2 : 0;     // K sub-offset for upper half-wave
  const int row0 = blockIdx.x * 128 + wave * 16;

  v8f acc[8] = {};

  const float* __restrict__ xrow = x + (size_t)(row0 + m) * C;
#pragma unroll 4
  for (int k = 0; k < C; k += 4) {
    v2f a;
    a.x = xrow[k + kh];
    a.y = xrow[k + kh + 1];
#pragma unroll
    for (int t = 0; t < 8; ++t) {
      const int col = t * 16 + m;           // output channel for this lane
      v2f bf;                               // B[k'][col] = W[col][k']
      bf.x = W[(size_t)col * C + k + kh];
      bf.y = W[(size_t)col * C + k + kh + 1];
      acc[t] = __builtin_amdgcn_wmma_f32_16x16x4_f32(
          /*neg_a=*/false, a, /*neg_b=*/false, bf,
          /*c_mod=*/(short)0, acc[t], /*reuse_a=*/false, /*reuse_b=*/false);
    }
  }

  const int rowA = row0 + ((lane >> 4) ? 8 : 0);
#pragma unroll
  for (int t = 0; t < 8; ++t) {
    const int n = t * 16 + m;
#pragma unroll
    for (int v = 0; v < 8; ++v) {
      xt[(size_t)(rowA + v) * C + n] = acc[t][v];
    }
  }
}

// ---------------------------------------------- temb_p = silu(temb) @ Wt^T + bt
__global__ __launch_bounds__(128) void temb_proj(const float* __restrict__ temb,
                                                 const float* __restrict__ Wt,
                                                 const float* __restrict__ bt,
                                                 float* __restrict__ temb_p) {
  const int o = threadIdx.x;  // 128 output channels
  float sum = bt[o];
  for (int t = 0; t < T_CH; ++t) {
    float v = temb[t];
    float s = v / (1.0f + __expf(-v));      // silu
    sum += s * Wt[(size_t)o * T_CH + t];
  }
  temb_p[o] = sum;
}

// -------------------------------- scatter pass 1: e[edge] += xt[node]*Binv[edge]
// one wave per nonzero; 32 lanes x float4 = 128 channels
__global__ __launch_bounds__(256) void scatter_node_to_edge(const float* __restrict__ xt,
                                                            const int* __restrict__ node_idx,
                                                            const int* __restrict__ edge_idx,
                                                            const float* __restrict__ Binv,
                                                            float* __restrict__ e, int nnz) {
  const int w    = (int)(((size_t)blockIdx.x * 256 + threadIdx.x) >> 5);
  const int lane = threadIdx.x & 31;
  if (w >= nnz) return;
  const int node = node_idx[w];
  const int edge = edge_idx[w];
  const float s  = Binv[edge];
  const float4 v = ((const float4*)(xt + (size_t)node * C))[lane];
  float* dst = e + (size_t)edge * C + lane * 4;
  atomicAdd(dst + 0, v.x * s);
  atomicAdd(dst + 1, v.y * s);
  atomicAdd(dst + 2, v.z * s);
  atomicAdd(dst + 3, v.w * s);
}

// -------------------------------- scatter pass 2: out[node] += e[edge]*Dinv[node]
__global__ __launch_bounds__(256) void scatter_edge_to_node(const float* __restrict__ e,
                                                            const int* __restrict__ node_idx,
                                                            const int* __restrict__ edge_idx,
                                                            const float* __restrict__ Dinv,
                                                            float* __restrict__ out, int nnz) {
  const int w    = (int)(((size_t)blockIdx.x * 256 + threadIdx.x) >> 5);
  const int lane = threadIdx.x & 31;
  if (w >= nnz) return;
  const int node = node_idx[w];
  const int edge = edge_idx[w];
  const float s  = Dinv[node];
  const float4 v = ((const float4*)(e + (size_t)edge * C))[lane];
  float* dst = out + (size_t)node * C + lane * 4;
  atomicAdd(dst + 0, v.x * s);
  atomicAdd(dst + 1, v.y * s);
  atomicAdd(dst + 2, v.z * s);
  atomicAdd(dst + 3, v.w * s);
}

// -------------------------------- finalize: h = silu(acc + b + temb_p), in place
__global__ __launch_bounds__(256) void finalize_kernel(float* __restrict__ out,
                                                       const float* __restrict__ b,
                                                       const float* __restrict__ temb_p,
                                                       size_t total) {
  size_t i = (size_t)blockIdx.x * 256 + threadIdx.x;
  if (i < total) {
    const int c = (int)(i & (C - 1));
    float v = out[i] + b[c] + temb_p[c];
    out[i] = v / (1.0f + __expf(-v));
  }
}

// --------------------------------------------------------------------------

extern "C" void kernel_launch(void* const* d_in, const int* in_sizes, int n_in,
                              void* d_out, int out_size, void* d_ws, size_t ws_size,
                              hipStream_t stream) {
  const float* x    = (const float*)d_in[0];
  const int*   hidx = (const int*)d_in[1];
  const float* temb = (const float*)d_in[2];
  const float* W    = (const float*)d_in[3];
  const float* b    = (const float*)d_in[4];
  const float* Wt   = (const float*)d_in[5];
  const float* bt   = (const float*)d_in[6];
  float*       out  = (float*)d_out;

  const int N   = in_sizes[0] / C;   // 131072
  const int NNZ = in_sizes[1] / 2;   // 2097152
  const int* node_idx = hidx;
  const int* edge_idx = hidx + NNZ;

  // workspace layout (floats)
  float* ws     = (float*)d_ws;
  float* xt     = ws;                                // N*C
  float* e      = xt + (size_t)N * C;                // E*C
  float* Dd     = e + (size_t)E_SEG * C;             // N  (degree -> Dinv in place)
  float* Bd     = Dd + N;                            // E  (edge size -> Binv in place)
  float* temb_p = Bd + E_SEG;                        // C

  // 1) zero all accumulators (must happen every call: graph replay reuses buffers)
  zero_f32<<<2048, 256, 0, stream>>>(e, (size_t)E_SEG * C);
  zero_f32<<<256, 256, 0, stream>>>(Dd, (size_t)N);
  zero_f32<<<256, 256, 0, stream>>>(Bd, (size_t)E_SEG);
  zero_f32<<<4096, 256, 0, stream>>>(out, (size_t)2 * N * C);  // h acc + res_out

  // 2) degrees, then reciprocals
  degree_kernel<<<(NNZ + 255) / 256, 256, 0, stream>>>(node_idx, edge_idx, Dd, Bd, NNZ);
  recip_kernel<<<(N + 255) / 256, 256, 0, stream>>>(Dd, N);
  recip_kernel<<<(E_SEG + 255) / 256, 256, 0, stream>>>(Bd, E_SEG);

  // 3) xt = x @ W^T via fp32 WMMA (16x16x4), 128 rows per block / 16 per wave
  gemm_xt_wmma<<<N / 128, 256, 0, stream>>>(x, W, xt, N);

  // 4) temb projection (tiny)
  temb_proj<<<1, C, 0, stream>>>(temb, Wt, bt, temb_p);

  // 5) node -> hyperedge scatter (L2-resident xt/e)
  scatter_node_to_edge<<<(NNZ + 7) / 8, 256, 0, stream>>>(xt, node_idx, edge_idx, Bd, e, NNZ);

  // 6) hyperedge -> node scatter, accumulating directly into d_out[0 : N*C]
  scatter_edge_to_node<<<(NNZ + 7) / 8, 256, 0, stream>>>(e, node_idx, edge_idx, Dd, out, NNZ);

  // 7) h = silu(acc + b + temb_p) in place; res_out already zeroed
  finalize_kernel<<<(unsigned)(((size_t)N * C + 255) / 256), 256, 0, stream>>>(
      out, b, temb_p, (size_t)N * C);
}